// SparseConv3dPack_74483322847791
// MI455X (gfx1250) — compile-verified
//
#include <hip/hip_runtime.h>

typedef float v2f __attribute__((ext_vector_type(2)));
typedef float v8f __attribute__((ext_vector_type(8)));

#define BB 2
#define CIN 64
#define COUT 64
#define DD 8
#define HH 48
#define WW 48
#define KPTS 27
#define SPB (DD*HH*WW)                 /* 18432 spatial per batch  */
#define NTOT (BB*SPB)                  /* 36864 total columns      */
#define KDIM (CIN*KPTS)                /* 1728 GEMM K              */
#define NKT (KDIM/4)                   /* 432 WMMA K steps (K=4)   */
#define TAPS_CH 9                      /* taps per LDS chunk       */
#define KT_CH (TAPS_CH*CIN/4)          /* 144 WMMA steps / chunk   */
#define LDS_FLOATS (TAPS_CH*CIN*16)    /* 9216 floats = 36 KB      */

/* workspace layout in floats */
#define XT_OFF   0
#define XT_SZ    (BB*SPB*CIN)          /* 2359296 channels-last x  */
#define OFFS_OFF (XT_OFF + XT_SZ)
#define OFFS_SZ  (BB*81*SPB)           /* 2985984 offset field     */
#define APKM_OFF (OFFS_OFF + OFFS_SZ)
#define APKM_SZ  (NKT*4*32*2)          /* 110592 packed main W     */
#define APKO_OFF (APKM_OFF + APKM_SZ)
#define APKO_SZ  (NKT*6*32*2)          /* 165888 packed offset W   */

__device__ __forceinline__ int iclamp(int v, int lo, int hi) {
    return v < lo ? lo : (v > hi ? hi : v);
}

/* ---------------- prep: transpose x + pack both A operands ---------------- */
__global__ __launch_bounds__(256) void dcn3d_prep(
    const float* __restrict__ x, const float* __restrict__ w_off,
    const float* __restrict__ w_main, float* __restrict__ ws)
{
    const long long e = (long long)blockIdx.x * 256 + threadIdx.x;
    float* xt   = ws + XT_OFF;
    float* apkm = ws + APKM_OFF;
    float* apko = ws + APKO_OFF;
    if (e < (long long)XT_SZ) {
        // x[b][c][sp] -> xt[b][sp][c]   (read contiguous)
        int sp = (int)(e % SPB);
        int t  = (int)(e / SPB);
        int ci = t % CIN;
        int b  = t / CIN;
        xt[((long long)b * SPB + sp) * CIN + ci] = x[e];
    } else if (e < (long long)XT_SZ + APKM_SZ) {
        // A layout for 16x16x4 f32: lane<16 -> M=lane, K=kt*4+{0,1};
        //                           lane>=16 -> M=lane-16, K=kt*4+{2,3}
        int e2   = (int)(e - XT_SZ);
        int p    = e2 & 1;
        int lane = (e2 >> 1) & 31;
        int mt   = (e2 >> 6) & 3;
        int kt   = e2 >> 8;
        int kk   = kt * 4 + ((lane >> 4) << 1) + p;   // GEMM-K index
        int ktap = kk >> 6;                           // tap (K-major: kk = tap*64+i)
        int ci   = kk & 63;
        int o    = mt * 16 + (lane & 15);
        apkm[e2] = w_main[((long long)o * CIN + ci) * KPTS + ktap];
    } else if (e < (long long)XT_SZ + APKM_SZ + APKO_SZ) {
        int e3   = (int)(e - XT_SZ - APKM_SZ);
        int p    = e3 & 1;
        int lane = (e3 >> 1) & 31;
        int mt   = (e3 >> 6) % 6;
        int kt   = (e3 >> 6) / 6;
        int kk   = kt * 4 + ((lane >> 4) << 1) + p;
        int ktap = kk >> 6;
        int ci   = kk & 63;
        int m    = mt * 16 + (lane & 15);
        apko[e3] = (m < 81) ? w_off[((long long)m * CIN + ci) * KPTS + ktap] : 0.0f;
    }
}

/* ------------- offset conv as implicit GEMM (M=96 padded, fp32 WMMA) ------ */
__global__ __launch_bounds__(128) void dcn3d_offset_conv(
    const float* __restrict__ xt, const float* __restrict__ apko,
    const float* __restrict__ cob, float* __restrict__ offs)
{
    __shared__ float vals[LDS_FLOATS];     // [tap*64+i][16]
    const int tid  = threadIdx.x;
    const int lane = tid & 31;
    const int wave = tid >> 5;
    const int i0   = tid >> 4;             // 0..7, channel octet
    const int nl   = tid & 15;             // column in tile
    const int n0   = blockIdx.x * 16;
    const int b    = n0 / SPB;
    const int sp0  = n0 % SPB;
    const int d    = sp0 / (HH * WW);
    const int rem  = sp0 % (HH * WW);
    const int h    = rem / WW;
    const int w0   = rem % WW;
    const float* xb = xt + (long long)b * SPB * CIN;

    v8f c0 = {};
    v8f c1 = {};
    const int mt0 = wave;                  // 0..3
    const int mt1 = wave + 4;              // 4,5 (waves 0,1 only)
    const int half = (lane >> 4) << 1;     // 0 or 2 (B K-rows)
    const int col  = lane & 15;

    for (int chunk = 0; chunk < 3; ++chunk) {
        // ---- im2col fill: 9 taps x 64 ch x 16 cols ----
        for (int kkl = 0; kkl < TAPS_CH; ++kkl) {
            const int k  = chunk * TAPS_CH + kkl;
            const int kd = k / 9, kh = (k / 3) % 3, kw = k % 3;
            const int zd = d + kd - 1;
            const int zh = h + kh - 1;
            const int zw = w0 + nl + kw - 1;
            float4 v0 = make_float4(0.f, 0.f, 0.f, 0.f);
            float4 v1 = v0;
            if (zd >= 0 && zd < DD && zh >= 0 && zh < HH && zw >= 0 && zw < WW) {
                const float4* p = (const float4*)(xb + (long long)((zd * HH + zh) * WW + zw) * CIN + i0 * 8);
                v0 = p[0]; v1 = p[1];
            }
            float* row = &vals[(kkl * CIN + i0 * 8) * 16 + nl];
            row[0*16] = v0.x; row[1*16] = v0.y; row[2*16] = v0.z; row[3*16] = v0.w;
            row[4*16] = v1.x; row[5*16] = v1.y; row[6*16] = v1.z; row[7*16] = v1.w;
        }
        __syncthreads();
        // ---- WMMA over this chunk (uniform branches, pipelined operands) ----
        const float* a0p = apko + ((long long)((chunk * KT_CH) * 6 + mt0) * 32 + lane) * 2;
        const float* bp  = vals + half * 16 + col;
        if (wave < 2) {
            const float* a1p = a0p + (long long)(mt1 - mt0) * 32 * 2;   // +4 tiles
            v2f a0n = *(const v2f*)a0p;
            v2f a1n = *(const v2f*)a1p;
            v2f bn; bn.x = bp[0]; bn.y = bp[16];
            #pragma unroll 4
            for (int ktl = 0; ktl < KT_CH; ++ktl) {
                v2f a0c = a0n, a1c = a1n, bc = bn;
                if (ktl + 1 < KT_CH) {
                    a0n = *(const v2f*)(a0p + (long long)(ktl + 1) * 6 * 32 * 2);
                    a1n = *(const v2f*)(a1p + (long long)(ktl + 1) * 6 * 32 * 2);
                    bn.x = bp[(ktl + 1) * 64];
                    bn.y = bp[(ktl + 1) * 64 + 16];
                }
                c0 = __builtin_amdgcn_wmma_f32_16x16x4_f32(false, a0c, false, bc, (short)0, c0, false, false);
                c1 = __builtin_amdgcn_wmma_f32_16x16x4_f32(false, a1c, false, bc, (short)0, c1, false, false);
            }
        } else {
            v2f a0n = *(const v2f*)a0p;
            v2f bn; bn.x = bp[0]; bn.y = bp[16];
            #pragma unroll 4
            for (int ktl = 0; ktl < KT_CH; ++ktl) {
                v2f a0c = a0n, bc = bn;
                if (ktl + 1 < KT_CH) {
                    a0n = *(const v2f*)(a0p + (long long)(ktl + 1) * 6 * 32 * 2);
                    bn.x = bp[(ktl + 1) * 64];
                    bn.y = bp[(ktl + 1) * 64 + 16];
                }
                c0 = __builtin_amdgcn_wmma_f32_16x16x4_f32(false, a0c, false, bc, (short)0, c0, false, false);
            }
        }
        __syncthreads();
    }
    // ---- epilogue: + bias, write first 81 rows ----
    const int rowoff = (lane >> 4) * 8;
    #pragma unroll
    for (int r = 0; r < 8; ++r) {
        int m = mt0 * 16 + rowoff + r;     // always < 64 < 81
        offs[((long long)(b * 81 + m)) * SPB + sp0 + col] = c0[r] + cob[m];
    }
    if (wave < 2) {
        #pragma unroll
        for (int r = 0; r < 8; ++r) {
            int m = mt1 * 16 + rowoff + r;
            if (m < 81)
                offs[((long long)(b * 81 + m)) * SPB + sp0 + col] = c1[r] + cob[m];
        }
    }
}

/* --------- fused trilinear gather + main GEMM (fp32 WMMA, LDS tile) ------- */
__global__ __launch_bounds__(128) void dcn3d_main(
    const float* __restrict__ xt, const float* __restrict__ offs,
    const float* __restrict__ apkm, const float* __restrict__ bias,
    float* __restrict__ out)
{
    __shared__ float vals[LDS_FLOATS];
    const int tid  = threadIdx.x;
    const int lane = tid & 31;
    const int wave = tid >> 5;
    const int i0   = tid >> 4;
    const int nl   = tid & 15;
    const int n0   = blockIdx.x * 16;
    const int b    = n0 / SPB;
    const int sp0  = n0 % SPB;
    const int d    = sp0 / (HH * WW);
    const int rem  = sp0 % (HH * WW);
    const int h    = rem / WW;
    const int w0   = rem % WW;
    const int sp   = sp0 + nl;
    const float* xb = xt + (long long)b * SPB * CIN;
    const float* ob = offs + (long long)b * 81 * SPB + sp;

    v8f c = {};
    const int mt   = wave;                 // Cout tile 0..3
    const int half = (lane >> 4) << 1;
    const int col  = lane & 15;

    for (int chunk = 0; chunk < 3; ++chunk) {
        // ---- trilinear gather into LDS ----
        for (int kkl = 0; kkl < TAPS_CH; ++kkl) {
            const int k  = chunk * TAPS_CH + kkl;
            const int kd = k / 9, kh = (k / 3) % 3, kw = k % 3;
            const float pd = ob[(long long)k * SPB]        + (float)(d + kd - 1);
            const float ph = ob[(long long)(27 + k) * SPB] + (float)(h + kh - 1);
            const float pw = ob[(long long)(54 + k) * SPB] + (float)(w0 + nl + kw - 1);
            const float fd0 = floorf(pd), fh0 = floorf(ph), fw0 = floorf(pw);
            const int dI = (int)fd0, hI = (int)fh0, wI = (int)fw0;
            const float td = pd - fd0, th = ph - fh0, tw = pw - fw0;
            float acc[8];
            #pragma unroll
            for (int j = 0; j < 8; ++j) acc[j] = 0.0f;
            #pragma unroll
            for (int corner = 0; corner < 8; ++corner) {
                const int cd = (corner >> 2) & 1, chh = (corner >> 1) & 1, cw = corner & 1;
                const int di = dI + cd, hi = hI + chh, wi = wI + cw;
                const bool valid = (di >= 0) && (di < DD) && (hi >= 0) && (hi < HH) &&
                                   (wi >= 0) && (wi < WW);
                float wgt = (cd ? td : 1.0f - td) * (chh ? th : 1.0f - th) *
                            (cw ? tw : 1.0f - tw);
                wgt = valid ? wgt : 0.0f;
                const int zidx = (iclamp(di, 0, DD - 1) * HH + iclamp(hi, 0, HH - 1)) * WW +
                                 iclamp(wi, 0, WW - 1);
                const float4* p = (const float4*)(xb + (long long)zidx * CIN + i0 * 8);
                const float4 v0 = p[0];
                const float4 v1 = p[1];
                acc[0] += wgt * v0.x; acc[1] += wgt * v0.y;
                acc[2] += wgt * v0.z; acc[3] += wgt * v0.w;
                acc[4] += wgt * v1.x; acc[5] += wgt * v1.y;
                acc[6] += wgt * v1.z; acc[7] += wgt * v1.w;
            }
            float* row = &vals[(kkl * CIN + i0 * 8) * 16 + nl];
            #pragma unroll
            for (int j = 0; j < 8; ++j) row[j * 16] = acc[j];
        }
        __syncthreads();
        // ---- WMMA over this chunk (pipelined operands) ----
        const float* ap = apkm + ((long long)((chunk * KT_CH) * 4 + mt) * 32 + lane) * 2;
        const float* bp = vals + half * 16 + col;
        v2f an = *(const v2f*)ap;
        v2f bn; bn.x = bp[0]; bn.y = bp[16];
        #pragma unroll 4
        for (int ktl = 0; ktl < KT_CH; ++ktl) {
            v2f ac = an, bc = bn;
            if (ktl + 1 < KT_CH) {
                an = *(const v2f*)(ap + (long long)(ktl + 1) * 4 * 32 * 2);
                bn.x = bp[(ktl + 1) * 64];
                bn.y = bp[(ktl + 1) * 64 + 16];
            }
            c = __builtin_amdgcn_wmma_f32_16x16x4_f32(false, ac, false, bc, (short)0, c, false, false);
        }
        __syncthreads();
    }
    // ---- epilogue: + bias ----
    const int rowoff = (lane >> 4) * 8;
    #pragma unroll
    for (int r = 0; r < 8; ++r) {
        const int o = mt * 16 + rowoff + r;
        out[((long long)(b * COUT + o)) * SPB + sp0 + col] = c[r] + bias[o];
    }
}

extern "C" void kernel_launch(void* const* d_in, const int* in_sizes, int n_in,
                              void* d_out, int out_size, void* d_ws, size_t ws_size,
                              hipStream_t stream) {
    const float* x     = (const float*)d_in[0];
    const float* w_off = (const float*)d_in[1];
    const float* b_off = (const float*)d_in[2];
    const float* w     = (const float*)d_in[3];
    const float* bias  = (const float*)d_in[4];
    float* out = (float*)d_out;
    float* ws  = (float*)d_ws;

    float* xt   = ws + XT_OFF;
    float* offs = ws + OFFS_OFF;
    float* apkm = ws + APKM_OFF;
    float* apko = ws + APKO_OFF;

    const int prep_total = XT_SZ + APKM_SZ + APKO_SZ;     // 2,635,776
    dcn3d_prep<<<(prep_total + 255) / 256, 256, 0, stream>>>(x, w_off, w, ws);

    const int ntiles = NTOT / 16;                          // 2304
    dcn3d_offset_conv<<<ntiles, 128, 0, stream>>>(xt, apko, b_off, offs);
    dcn3d_main<<<ntiles, 128, 0, stream>>>(xt, offs, apkm, bias, out);
}